// CenterLoss_62191126446263
// MI455X (gfx1250) — compile-verified
//
#include <hip/hip_runtime.h>
#include <hip/hip_bf16.h>
#include <math.h>

// CenterLoss, exploiting the one-hot structure:
//   loss = ( sum_b clamp(cos(x_b, centers[label_b]), 1e-12, 1e12)
//            + (B*C - B)*1e-12 ) / B
// Gathered dot products computed on the WMMA unit (V_WMMA_F32_16X16X4_F32,
// full fp32) via the diagonal of a 16x16 tile.

typedef __attribute__((ext_vector_type(2))) float v2f;
typedef __attribute__((ext_vector_type(8))) float v8f;

#define B_ROWS 1024
#define D_DIM  512
#define C_CENT 100000
#define EPS_F       1e-8f
#define CLAMP_MIN_F 1e-12f
#define CLAMP_MAX_F 1e12f

// One wave (32 lanes) per block; each block handles 16 rows of the batch.
// WMMA f32 16x16x4 fragment layout (ISA 7.12.2):
//   A (16x4):  lane m (0-15): VGPR0=A[m][k0], VGPR1=A[m][k0+1]
//              lane m+16:     VGPR0=A[m][k0+2], VGPR1=A[m][k0+3]
//   B (4x16):  lane n (0-15): VGPR0=B[k0][n],  VGPR1=B[k0+1][n]
//              lane n+16:     VGPR0=B[k0+2][n], VGPR1=B[k0+3][n]
// => every lane loads one contiguous float2 from its row for both fragments.
__global__ __launch_bounds__(32)
void center_loss_rows(const float* __restrict__ input,
                      const int*   __restrict__ label,
                      const float* __restrict__ centers,
                      float*       __restrict__ row_vals) {
    const int lane  = threadIdx.x;            // 0..31 (wave32)
    const int row16 = lane & 15;              // row within the 16-row tile
    const int khalf = (lane >> 4) << 1;       // 0 for lanes 0-15, 2 for 16-31
    const int grow  = blockIdx.x * 16 + row16;

    const float* arow = input + (size_t)grow * D_DIM;
    const int    lab  = label[grow];
    const float* brow = centers + (size_t)lab * D_DIM;

    v8f acc = {};                 // 16x16 fp32 accumulator tile
    float na = 0.0f, nb = 0.0f;   // partial ||x||^2, ||c||^2 (this lane's K slice)

#pragma unroll 4
    for (int k0 = 0; k0 < D_DIM; k0 += 4) {
        v2f a = *(const v2f*)(arow + k0 + khalf);
        v2f b = *(const v2f*)(brow + k0 + khalf);
        na = fmaf(a.x, a.x, fmaf(a.y, a.y, na));
        nb = fmaf(b.x, b.x, fmaf(b.y, b.y, nb));
        // D = A*B + C, fp32 in/out
        acc = __builtin_amdgcn_wmma_f32_16x16x4_f32(
                  /*neg_a=*/false, a, /*neg_b=*/false, b,
                  /*c_mod=*/(short)0, acc,
                  /*reuse_a=*/false, /*reuse_b=*/false);
    }

    // Stage accumulators + norm partials through LDS to extract the diagonal.
    __shared__ float s_acc[32][8];
    __shared__ float s_pa[32];
    __shared__ float s_pb[32];
#pragma unroll
    for (int j = 0; j < 8; ++j) s_acc[lane][j] = acc[j];
    s_pa[lane] = na;
    s_pb[lane] = nb;
    __syncthreads();

    if (lane < 16) {
        // Diagonal element (m,m): m<8 -> lane m, acc[m]; m>=8 -> lane m+16, acc[m-8]
        const int srcLane = (lane < 8) ? lane : (lane + 16);
        const float ip = s_acc[srcLane][lane & 7];
        const float w1 = sqrtf(s_pa[lane] + s_pa[lane + 16]);  // ||input_row||
        const float w2 = sqrtf(s_pb[lane] + s_pb[lane + 16]);  // ||center_row||
        const float denom = fmaxf(w1 * w2, EPS_F);
        const float cosv  = ip / denom;
        row_vals[blockIdx.x * 16 + lane] =
            fminf(fmaxf(cosv, CLAMP_MIN_F), CLAMP_MAX_F);
    }
}

// Deterministic single-block reduction of the 1024 per-row values.
__global__ __launch_bounds__(256)
void center_loss_reduce(const float* __restrict__ row_vals,
                        float*       __restrict__ out) {
    __shared__ float s[256];
    const int t = threadIdx.x;
    float v = row_vals[t] + row_vals[t + 256] +
              row_vals[t + 512] + row_vals[t + 768];
    s[t] = v;
    __syncthreads();
#pragma unroll
    for (int off = 128; off > 0; off >>= 1) {
        if (t < off) s[t] += s[t + off];
        __syncthreads();
    }
    if (t == 0) {
        // All (B*C - B) masked zeros clamp to 1e-12 and survive the sum.
        const float zero_term =
            (float)((double)B_ROWS * (double)(C_CENT - 1) * 1e-12);
        out[0] = (s[0] + zero_term) * (1.0f / (float)B_ROWS);
    }
}

extern "C" void kernel_launch(void* const* d_in, const int* in_sizes, int n_in,
                              void* d_out, int out_size, void* d_ws, size_t ws_size,
                              hipStream_t stream) {
    const float* input   = (const float*)d_in[0];   // [B, D] fp32
    const int*   label   = (const int*)  d_in[1];   // [B] int
    const float* centers = (const float*)d_in[2];   // [C, D] fp32
    float* row_vals = (float*)d_ws;                 // 1024 floats scratch
    float* out      = (float*)d_out;                // scalar loss

    center_loss_rows<<<B_ROWS / 16, 32, 0, stream>>>(input, label, centers,
                                                     row_vals);
    center_loss_reduce<<<1, 256, 0, stream>>>(row_vals, out);
}